// FurthestPointSampling_16346645529140
// MI455X (gfx1250) — compile-verified
//
#include <hip/hip_runtime.h>
#include <stdint.h>

// ---------------- problem constants (fixed by setup_inputs) ----------------
#define NB       16          // batches
#define NPTS     262144      // points per batch
#define NPOINT   1024        // samples
#define SEG      8           // blocks per batch
#define PTS      (NPTS / SEG)      // 32768 points per block
#define THREADS  256
#define WAVES    (THREADS / 32)
#define TILE     1024              // points per DMA tile
#define NT       (PTS / TILE)      // 32 tiles per round
#define TILE_B   (TILE * 12)       // 12288 bytes per tile (3 f32 / point)
#define DIST_B   (PTS * 4)         // 131072 B resident distance array
#define STAGE_B  (2 * TILE_B)      // 24576 B double buffer
#define LDS_BYTES (DIST_B + STAGE_B + 64 + WAVES * 8)  // + cvec pad + reduction scratch

typedef unsigned int  u32x4 __attribute__((ext_vector_type(4)));
typedef int           i32x8 __attribute__((ext_vector_type(8)));
typedef int           i32x4 __attribute__((ext_vector_type(4)));

__device__ __forceinline__ unsigned lds_off(const void* p) {
  // addrspace(3) -> generic keeps the LDS byte offset in the low 32 bits
  return (unsigned)(unsigned long long)p;
}

// One TDM descriptor: contiguous 12288-byte (3072 dword) tile, global -> LDS.
__device__ __forceinline__ void tdm_tile_load(unsigned ldsAddr, unsigned long long gaddr) {
  u32x4 g0;
  g0.x = 1u;                                                     // count=1, user desc
  g0.y = ldsAddr;                                                // lds_addr (bytes)
  g0.z = (unsigned)(gaddr & 0xffffffffull);                      // global_addr[31:0]
  g0.w = (unsigned)((gaddr >> 32) & 0x01ffffffull) | 0x80000000u; // addr[56:32] | type=2
  i32x8 g1;
  g1[0] = (int)(2u << 16);      // workgroup_mask=0, data_size=4B, no pad/iterate/barrier
  g1[1] = (int)(3072u << 16);   // tensor_dim0[15:0]=3072 (bits 63:48)
  g1[2] = (int)(1u << 16);      // tensor_dim0[31:16]=0 | tensor_dim1=1
  g1[3] = (int)(3072u << 16);   // tensor_dim1 hi=0 | tile_dim0=3072
  g1[4] = 0;                    // tile_dim1=0 (unused), tile_dim2=0
  g1[5] = 3072;                 // tensor_dim0_stride
  g1[6] = 0;
  g1[7] = 0;
  i32x4 z4; z4[0] = 0; z4[1] = 0; z4[2] = 0; z4[3] = 0;          // groups 2/3 unused (<=2D)
  i32x8 z8; z8[0] = 0; z8[1] = 0; z8[2] = 0; z8[3] = 0;
            z8[4] = 0; z8[5] = 0; z8[6] = 0; z8[7] = 0;
  __builtin_amdgcn_tensor_load_to_lds(g0, g1, z4, z4, z8, 0);
}

__global__ __launch_bounds__(THREADS)
void fps_kernel(const float* __restrict__ xyz, int* __restrict__ out,
                unsigned char* __restrict__ ws)
{
  extern __shared__ unsigned char smem[];
  float* dist  = (float*)smem;                                   // [PTS] resident
  float* stage = (float*)(smem + DIST_B);                        // [2][TILE*3]
  float* cvec  = (float*)(smem + DIST_B + STAGE_B);              // 3 floats
  float* wmaxv = (float*)(smem + DIST_B + STAGE_B + 64);         // [WAVES]
  int*   wmaxi = (int*)  (smem + DIST_B + STAGE_B + 64 + WAVES * 4);

  const int tid  = threadIdx.x;
  const int lane = tid & 31;
  const int wav  = tid >> 5;
  const int b    = blockIdx.x / SEG;
  const int s    = blockIdx.x % SEG;
  const int p0   = s * PTS;                                      // batch-local base point

  unsigned long long* partials = (unsigned long long*)ws;        // [NB][SEG] packed keys
  unsigned* arrive = (unsigned*)(ws + 1024) + b * 16;            // 64B-strided counter
  unsigned* bc     = (unsigned*)(ws + 2048) + b * 16;            // tag, idx, cx, cy, cz

  const unsigned long long gsegBase =
      (unsigned long long)xyz +
      ((unsigned long long)b * NPTS + (unsigned long long)p0) * 12ull;
  const unsigned stageOff = lds_off(stage);

  // distances start at "infinity"; each thread owns j == tid (mod 256)
  #pragma unroll 4
  for (int k = 0; k < PTS / THREADS; ++k)
    dist[tid + k * THREADS] = 1e10f;

  if (s == 0 && tid == 0) out[b * NPOINT] = 0;                   // emitted index 0

  for (int r = 1; r < NPOINT; ++r) {
    // ---- obtain this round's centroid (index published by previous round) ----
    if (tid == 0) {
      float cx, cy, cz;
      if (r == 1) {
        const float* cp = xyz + (unsigned long long)b * NPTS * 3ull;
        cx = cp[0]; cy = cp[1]; cz = cp[2];
      } else {
        while (__hip_atomic_load(&bc[0], __ATOMIC_ACQUIRE, __HIP_MEMORY_SCOPE_AGENT)
               < (unsigned)(r - 1))
          __builtin_amdgcn_s_sleep(1);
        cx = __uint_as_float(__hip_atomic_load(&bc[2], __ATOMIC_RELAXED, __HIP_MEMORY_SCOPE_AGENT));
        cy = __uint_as_float(__hip_atomic_load(&bc[3], __ATOMIC_RELAXED, __HIP_MEMORY_SCOPE_AGENT));
        cz = __uint_as_float(__hip_atomic_load(&bc[4], __ATOMIC_RELAXED, __HIP_MEMORY_SCOPE_AGENT));
      }
      cvec[0] = cx; cvec[1] = cy; cvec[2] = cz;
    }
    __syncthreads();
    const float cx = cvec[0], cy = cvec[1], cz = cvec[2];

    float bv = -1.0f;
    int   bj = 0;

    // ---- stream xyz tiles via TDM, double-buffered; update resident distances ----
    if (wav == 0) tdm_tile_load(stageOff, gsegBase);             // prime tile 0 -> buf 0

    for (int t = 0; t < NT; ++t) {
      if (t + 1 < NT) {
        if (wav == 0)
          tdm_tile_load(stageOff + ((unsigned)(t + 1) & 1u) * TILE_B,
                        gsegBase + (unsigned long long)(t + 1) * TILE_B);
        __builtin_amdgcn_s_wait_tensorcnt(1);                    // tile t landed
      } else {
        __builtin_amdgcn_s_wait_tensorcnt(0);
      }
      __syncthreads();                                           // tile t visible to all waves

      const float* tb = stage + (t & 1) * (TILE * 3);
      #pragma unroll
      for (int q = 0; q < TILE / THREADS; ++q) {
        const int l = tid + q * THREADS;                         // 0..TILE-1
        const int j = t * TILE + l;                              // block-local point
        const float dx = __fsub_rn(tb[l * 3 + 0], cx);
        const float dy = __fsub_rn(tb[l * 3 + 1], cy);
        const float dz = __fsub_rn(tb[l * 3 + 2], cz);
        // no FMA contraction: match mul-then-add of the reference
        const float d  = __fadd_rn(__fadd_rn(__fmul_rn(dx, dx), __fmul_rn(dy, dy)),
                                   __fmul_rn(dz, dz));
        const float nd = fminf(dist[j], d);
        dist[j] = nd;
        if (nd > bv) { bv = nd; bj = j; }                        // strict >: first max wins
      }
      __syncthreads();                                           // done with buf before reuse
    }

    // ---- wave32 argmax reduction (max value, ties -> lowest index) ----
    #pragma unroll
    for (int off = 16; off > 0; off >>= 1) {
      const float ov = __shfl_xor(bv, off, 32);
      const int   oj = __shfl_xor(bj, off, 32);
      if (ov > bv || (ov == bv && oj < bj)) { bv = ov; bj = oj; }
    }
    if (lane == 0) { wmaxv[wav] = bv; wmaxi[wav] = bj; }
    __syncthreads();

    if (tid == 0) {
      float fv = wmaxv[0]; int fj = wmaxi[0];
      #pragma unroll
      for (int w = 1; w < WAVES; ++w) {
        const float ov = wmaxv[w]; const int oj = wmaxi[w];
        if (ov > fv || (ov == fv && oj < fj)) { fv = ov; fj = oj; }
      }
      const unsigned gidx = (unsigned)(p0 + fj);                 // batch-local index
      const unsigned long long key =
          ((unsigned long long)__float_as_uint(fv) << 32) | (unsigned)(~gidx);
      __hip_atomic_store(&partials[b * SEG + s], key,
                         __ATOMIC_RELEASE, __HIP_MEMORY_SCOPE_AGENT);
      const unsigned old = __hip_atomic_fetch_add(arrive, 1u,
                         __ATOMIC_ACQ_REL, __HIP_MEMORY_SCOPE_AGENT);
      if (old == (unsigned)(r * SEG - 1)) {                      // last arrival of round r
        unsigned long long bestK = 0ull;
        #pragma unroll
        for (int q2 = 0; q2 < SEG; ++q2) {
          const unsigned long long k2 = __hip_atomic_load(&partials[b * SEG + q2],
                               __ATOMIC_RELAXED, __HIP_MEMORY_SCOPE_AGENT);
          if (k2 > bestK) bestK = k2;
        }
        const unsigned idx = ~(unsigned)bestK;                   // batch-local argmax
        out[b * NPOINT + r] = (int)idx;
        const float* cp = xyz + ((unsigned long long)b * NPTS + idx) * 3ull;
        __hip_atomic_store(&bc[2], __float_as_uint(cp[0]), __ATOMIC_RELAXED, __HIP_MEMORY_SCOPE_AGENT);
        __hip_atomic_store(&bc[3], __float_as_uint(cp[1]), __ATOMIC_RELAXED, __HIP_MEMORY_SCOPE_AGENT);
        __hip_atomic_store(&bc[4], __float_as_uint(cp[2]), __ATOMIC_RELAXED, __HIP_MEMORY_SCOPE_AGENT);
        __hip_atomic_store(&bc[1], idx,                    __ATOMIC_RELAXED, __HIP_MEMORY_SCOPE_AGENT);
        __hip_atomic_store(&bc[0], (unsigned)r,            __ATOMIC_RELEASE, __HIP_MEMORY_SCOPE_AGENT);
      }
    }
    // next round's leading __syncthreads keeps everyone behind tid0's publish/spin
  }
}

extern "C" void kernel_launch(void* const* d_in, const int* in_sizes, int n_in,
                              void* d_out, int out_size, void* d_ws, size_t ws_size,
                              hipStream_t stream) {
  (void)in_sizes; (void)n_in; (void)out_size; (void)ws_size;
  const float* xyz = (const float*)d_in[0];
  int* out = (int*)d_out;

  // allow >64KB dynamic LDS (idempotent, not a stream op -> capture-safe)
  (void)hipFuncSetAttribute(reinterpret_cast<const void*>(fps_kernel),
                            hipFuncAttributeMaxDynamicSharedMemorySize, (int)LDS_BYTES);

  // reset arrival counters / tags / partials every launch (deterministic)
  (void)hipMemsetAsync(d_ws, 0, 4096, stream);

  fps_kernel<<<dim3(NB * SEG), dim3(THREADS), LDS_BYTES, stream>>>(
      xyz, out, (unsigned char*)d_ws);
}